// GNN_layer_73083163508962
// MI455X (gfx1250) — compile-verified
//
#include <hip/hip_runtime.h>

typedef __attribute__((ext_vector_type(2))) float v2f;
typedef __attribute__((ext_vector_type(8))) float v8f;
typedef __attribute__((ext_vector_type(4))) int v4i;

#define NB 4
#define AIN 8
#define AOUT 16
#define XIN 64
#define XOUT 64
#define NN 1024
#define CSTRIDE (NN * NN)

// workspace layout (float offsets)
#define MOC_OFF   0          // [b][i][c]   4*1024*8
#define DP_OFF    32768      // [b][i][c]
#define MDP_OFF   65536      // [b][c]
#define MALL_OFF  65568      // [b][c]
#define MX_OFF    65600      // [b][k]      4*64
#define RT_OFF    65856      // [b][o][i]   4*16*1024  (R' transposed, S/2 folded in)
#define X1T_OFF   131392     // [b][i][u]   4*1024*64
#define X2T_OFF   393536     // [b][i][u]

// ---- CDNA5 async global->LDS path (guarded: falls back to plain copy) ----
#if defined(__has_builtin)
#if __has_builtin(__builtin_amdgcn_global_load_async_to_lds_b128) && \
    __has_builtin(__builtin_amdgcn_s_wait_asynccnt)
#define HAVE_ASYNC_LDS 1
#endif
#endif

typedef __attribute__((address_space(1))) v4i gv4i;   // global 16B granule
typedef __attribute__((address_space(3))) v4i lv4i;   // LDS 16B granule

// ---------------- K1: row means of A + diagonal extraction ----------------
__global__ void k1_rowstats(const float* __restrict__ A, float* __restrict__ ws) {
  int wv = threadIdx.x >> 5, lane = threadIdx.x & 31;
  int r = blockIdx.x * 8 + wv;            // row id over (b,c,i)
  int b = r >> 13, c = (r >> 10) & 7, i = r & 1023;
  const float* row = A + (size_t)((b * AIN + c) * NN + i) * NN;
  float s = 0.f;
  const float4* row4 = reinterpret_cast<const float4*>(row);
#pragma unroll
  for (int it = 0; it < 8; ++it) {
    float4 v = row4[it * 32 + lane];
    s += v.x + v.y + v.z + v.w;
  }
#pragma unroll
  for (int off = 16; off > 0; off >>= 1) s += __shfl_xor(s, off, 32);
  if (lane == 0) {
    ws[MOC_OFF + (b * NN + i) * AIN + c] = s * (1.f / 1024.f);
    ws[DP_OFF + (b * NN + i) * AIN + c] = row[i];
  }
}

// ---------------- K2a: per-batch means (tiny) ----------------
__global__ void k2a_means(const float* __restrict__ X, float* __restrict__ ws) {
  int b = blockIdx.x, t = threadIdx.x;
  if (t < 8) {
    float s = 0.f;
    for (int i = 0; i < NN; ++i) s += ws[DP_OFF + (b * NN + i) * AIN + t];
    ws[MDP_OFF + b * AIN + t] = s * (1.f / 1024.f);
  } else if (t < 16) {
    int c = t - 8;
    float s = 0.f;
    for (int i = 0; i < NN; ++i) s += ws[MOC_OFF + (b * NN + i) * AIN + c];
    ws[MALL_OFF + b * AIN + c] = s * (1.f / 1024.f);
  } else if (t < 80) {
    int k = t - 16;
    float s = 0.f;
    for (int i = 0; i < NN; ++i) s += X[(b * NN + i) * XIN + k];
    ws[MX_OFF + b * XIN + k] = s * (1.f / 1024.f);
  }
}

// ---------------- K2b: R'[b][o][i] = moc@W3 + X@W6 + dp@W5 + S/2 ----------------
__global__ void k2b_R(const float* __restrict__ X, const float* __restrict__ W2,
                      const float* __restrict__ W3, const float* __restrict__ W4,
                      const float* __restrict__ W5, const float* __restrict__ W6,
                      const float* __restrict__ W7, const float* __restrict__ bias,
                      float* __restrict__ ws) {
  int id = blockIdx.x * 256 + threadIdx.x;   // (b,i,o) : 4*1024*16
  int b = id >> 14, i = (id >> 4) & 1023, o = id & 15;
  const float* moc = ws + MOC_OFF + (b * NN + i) * AIN;
  const float* dp  = ws + DP_OFF + (b * NN + i) * AIN;
  const float* mal = ws + MALL_OFF + b * AIN;
  const float* mdp = ws + MDP_OFF + b * AIN;
  const float* mx  = ws + MX_OFF + b * XIN;
  const float* xi  = X + (size_t)(b * NN + i) * XIN;
  float r = 0.f, s = bias[o];
#pragma unroll
  for (int c = 0; c < AIN; ++c) {
    r += moc[c] * W3[c * AOUT + o] + dp[c] * W5[c * AOUT + o];
    s += mal[c] * W2[c * AOUT + o] + mdp[c] * W4[c * AOUT + o];
  }
  for (int k = 0; k < XIN; ++k) {
    r += xi[k] * W6[k * AOUT + o];
    s += mx[k] * W7[k * AOUT + o];
  }
  ws[RT_OFF + (b * AOUT + o) * NN + i] = r + 0.5f * s;
}

// ---------------- K2c: X1t / X2t branches ----------------
__global__ void k2c_xbranch(const float* __restrict__ X,
    const float* __restrict__ aW1, const float* __restrict__ aW2, const float* __restrict__ aW3,
    const float* __restrict__ aW4, const float* __restrict__ aW5, const float* __restrict__ aW6,
    const float* __restrict__ ab,
    const float* __restrict__ cW1, const float* __restrict__ cW2, const float* __restrict__ cW3,
    const float* __restrict__ cW4, const float* __restrict__ cW5, const float* __restrict__ cW6,
    const float* __restrict__ cb,
    float* __restrict__ ws) {
  int id = blockIdx.x * 256 + threadIdx.x;   // (b,i,u) : 4*1024*64
  int b = id >> 16, i = (id >> 6) & 1023, u = id & 63;
  const float* xi  = X + (size_t)(b * NN + i) * XIN;
  const float* mx  = ws + MX_OFF + b * XIN;
  const float* moc = ws + MOC_OFF + (b * NN + i) * AIN;
  const float* dp  = ws + DP_OFF + (b * NN + i) * AIN;
  const float* mal = ws + MALL_OFF + b * AIN;
  const float* mdp = ws + MDP_OFF + b * AIN;
  float x1 = ab[u], x2 = cb[u];
  for (int k = 0; k < XIN; ++k) {
    x1 += xi[k] * aW1[k * XOUT + u] + mx[k] * aW2[k * XOUT + u];
    x2 += xi[k] * cW1[k * XOUT + u] + mx[k] * cW2[k * XOUT + u];
  }
#pragma unroll
  for (int c = 0; c < AIN; ++c) {
    x1 += moc[c] * aW3[c * XOUT + u] + mal[c] * aW6[c * XOUT + u] +
          dp[c] * aW4[c * XOUT + u] + mdp[c] * aW5[c * XOUT + u];
    x2 += moc[c] * cW3[c * XOUT + u] + mal[c] * cW6[c * XOUT + u] +
          dp[c] * cW4[c * XOUT + u] + mdp[c] * cW5[c * XOUT + u];
  }
  ws[X1T_OFF + (b * NN + i) * XOUT + u] = x1;
  ws[X2T_OFF + (b * NN + i) * XOUT + u] = x2;
}

// ---------------- K3: At = W1^T x A-strip + R'_i + R'_j, stored as [b][o][i][j] ----------------
// WMMA mapping: M=o(16), N=j(16), K=c(4 per op, 2 ops for AIN=8). D VGPR p holds a
// contiguous 16-j row for o=p (lanes 0-15) / o=p+8 (lanes 16-31) -> coalesced stores.
// R'^T[b] (16x1024 = 64KB) is staged into LDS (async B128 where available) so the
// per-tile R'_j reads come off the DS pipe instead of VMEM.
__global__ void k3_At(const float* __restrict__ A, const float* __restrict__ W1,
                      const float* __restrict__ ws, float* __restrict__ outA) {
  extern __shared__ float ldsRt[];          // 16384 floats: R'^T[b] as [o][j]
  int wv = threadIdx.x >> 5, lane = threadIdx.x & 31;
  int half = lane >> 4, l = lane & 15;
  int b = blockIdx.x >> 7;
  int i = ((blockIdx.x & 127) << 3) + wv;

  const float* Rtb = ws + RT_OFF + (size_t)b * AOUT * NN;
  // cooperative stage of R'^T[b] into LDS: 4096 float4, 16 per thread
#ifdef HAVE_ASYNC_LDS
#pragma unroll
  for (int it = 0; it < 16; ++it) {
    int e = (it * 256 + (int)threadIdx.x) * 4;     // float index, 16B granules
    __builtin_amdgcn_global_load_async_to_lds_b128(
        (gv4i*)(Rtb + e), (lv4i*)(ldsRt + e), 0, 0);
  }
  __builtin_amdgcn_s_wait_asynccnt(0);
#else
#pragma unroll
  for (int it = 0; it < 16; ++it) {
    int e = it * 256 + (int)threadIdx.x;
    reinterpret_cast<float4*>(ldsRt)[e] = reinterpret_cast<const float4*>(Rtb)[e];
  }
#endif
  __syncthreads();

  // A-operand: W1^T (rows o, cols c). lanes<16: K={c0,c0+1}; lanes>=16: K={c0+2,c0+3}
  v2f a0, a1;
  a0.x = W1[(0 + 2 * half) * AOUT + l];
  a0.y = W1[(1 + 2 * half) * AOUT + l];
  a1.x = W1[(4 + 2 * half) * AOUT + l];
  a1.y = W1[(5 + 2 * half) * AOUT + l];

  float ri[8];
#pragma unroll
  for (int p = 0; p < 8; ++p) ri[p] = ldsRt[(p + 8 * half) * NN + i];

  const float* Ab = A + (size_t)b * AIN * CSTRIDE + (size_t)i * NN;
  float* Ob = outA + (size_t)b * AOUT * CSTRIDE + (size_t)i * NN;

  for (int j0 = 0; j0 < NN; j0 += 16) {
    int j = j0 + l;
    // B-operand: A data, rows K=c, cols N=j. v0 = rows {c0, c0+2}, v1 = rows {c0+1, c0+3}
    const float* p0 = Ab + (size_t)(0 + 2 * half) * CSTRIDE + j;
    const float* p1 = Ab + (size_t)(1 + 2 * half) * CSTRIDE + j;
    const float* p2 = Ab + (size_t)(4 + 2 * half) * CSTRIDE + j;
    const float* p3 = Ab + (size_t)(5 + 2 * half) * CSTRIDE + j;
    v2f b0, b1;
    b0.x = p0[0];
    b0.y = p1[0];
    b1.x = p2[0];
    b1.y = p3[0];
    // prefetch the four A-plane streams ~2KB ahead; streaming data used once ->
    // locality 0 (lowered SYS scope: GL2 only, keeps WGP$ clean)
    __builtin_prefetch(p0 + 512, 0, 0);
    __builtin_prefetch(p1 + 512, 0, 0);
    __builtin_prefetch(p2 + 512, 0, 0);
    __builtin_prefetch(p3 + 512, 0, 0);
    v8f d = {};
    d = __builtin_amdgcn_wmma_f32_16x16x4_f32(false, a0, false, b0, (short)0, d, false, false);
    d = __builtin_amdgcn_wmma_f32_16x16x4_f32(false, a1, false, b1, (short)0, d, false, false);
#pragma unroll
    for (int p = 0; p < 8; ++p) {
      float rj = ldsRt[(p + 8 * half) * NN + j];
      Ob[(size_t)(p + 8 * half) * CSTRIDE + j] = d[p] + ri[p] + rj;
    }
  }
}

// ---------------- K4: fused einsum (At^T[b,s] @ X2t / n) + 1024->64 projection ----------------
// 16 waves per WG; wave s computes out_pre[i-tile, s, 0:64] with K=j WMMA loop, stages it
// in LDS ([k][i] layout, 64KB), then 4 waves do the (s,t)->u GEMM, add bias + X1t, store.
__global__ void k4_out(const float* __restrict__ Aout, const float* __restrict__ outW,
                       const float* __restrict__ outB, const float* __restrict__ ws,
                       float* __restrict__ out2) {
  extern __shared__ float lds[];   // 16384 floats: out_pre [k=1024][i=16]
  int wv = threadIdx.x >> 5, lane = threadIdx.x & 31;
  int half = lane >> 4, l = lane & 15;
  int b = blockIdx.x >> 6;
  int i0 = (blockIdx.x & 63) << 4;
  int s = wv;  // 0..15

  const float* Ap = Aout + (size_t)(b * AOUT + s) * CSTRIDE + (size_t)(i0 + l) * NN;
  const float* X2 = ws + X2T_OFF + (size_t)b * NN * XOUT;

  v8f acc[4] = {};
  for (int k0 = 0; k0 < NN; k0 += 4) {
    // A-operand: rows i (both halves same rows), lanes<16 K={k0,k0+1}, lanes>=16 K={k0+2,k0+3}
    v2f a;
    const float* ap = Ap + k0 + 2 * half;
    a.x = ap[0];
    a.y = ap[1];
    // strided row stream re-uses each 128B line over 8 K-steps: prefetch into
    // ALL levels (locality 3 -> WGP scope) so L0 covers the scattered loads
    __builtin_prefetch(ap + 512, 0, 3);
#pragma unroll
    for (int tt = 0; tt < 4; ++tt) {
      v2f bb;
      bb.x = X2[(k0 + 2 * half) * XOUT + tt * 16 + l];
      bb.y = X2[(k0 + 1 + 2 * half) * XOUT + tt * 16 + l];
      acc[tt] = __builtin_amdgcn_wmma_f32_16x16x4_f32(false, a, false, bb, (short)0,
                                                      acc[tt], false, false);
    }
  }
  // stage out_pre (scaled by 1/n) into LDS: index k = s*64 + t, layout [k][16 i]
#pragma unroll
  for (int tt = 0; tt < 4; ++tt)
#pragma unroll
    for (int p = 0; p < 8; ++p)
      lds[(s * 64 + tt * 16 + l) * 16 + p + 8 * half] = acc[tt][p] * (1.f / 1024.f);
  __syncthreads();

  if (wv < 4) {  // wave-uniform: EXEC all ones inside
    int u0 = wv * 16;
    v8f f = {};
    for (int k0 = 0; k0 < AOUT * XOUT; k0 += 4) {
      v2f a, bb;
      a.x = lds[(k0 + 2 * half) * 16 + l];
      a.y = lds[(k0 + 2 * half + 1) * 16 + l];
      bb.x = outW[(k0 + 2 * half) * XOUT + u0 + l];
      bb.y = outW[(k0 + 2 * half + 1) * XOUT + u0 + l];
      f = __builtin_amdgcn_wmma_f32_16x16x4_f32(false, a, false, bb, (short)0, f, false, false);
    }
    float bu = outB[u0 + l];
    const float* X1 = ws + X1T_OFF + (size_t)b * NN * XOUT;
#pragma unroll
    for (int p = 0; p < 8; ++p) {
      int i = i0 + p + 8 * half;
      out2[(size_t)(b * NN + i) * XOUT + u0 + l] = f[p] + bu + X1[i * XOUT + u0 + l];
    }
  }
}

extern "C" void kernel_launch(void* const* d_in, const int* in_sizes, int n_in,
                              void* d_out, int out_size, void* d_ws, size_t ws_size,
                              hipStream_t stream) {
  const float* A    = (const float*)d_in[0];
  const float* X    = (const float*)d_in[1];
  const float* A_W1 = (const float*)d_in[2];
  const float* A_W2 = (const float*)d_in[3];
  const float* A_W3 = (const float*)d_in[4];
  const float* A_W4 = (const float*)d_in[5];
  const float* A_W5 = (const float*)d_in[6];
  const float* A_W6 = (const float*)d_in[7];
  const float* A_W7 = (const float*)d_in[8];
  const float* A_b  = (const float*)d_in[9];
  const float* X1W1 = (const float*)d_in[10];
  const float* X1W2 = (const float*)d_in[11];
  const float* X1W3 = (const float*)d_in[12];
  const float* X1W4 = (const float*)d_in[13];
  const float* X1W5 = (const float*)d_in[14];
  const float* X1W6 = (const float*)d_in[15];
  const float* X1b  = (const float*)d_in[16];
  const float* X2W1 = (const float*)d_in[17];
  const float* X2W2 = (const float*)d_in[18];
  const float* X2W3 = (const float*)d_in[19];
  const float* X2W4 = (const float*)d_in[20];
  const float* X2W5 = (const float*)d_in[21];
  const float* X2W6 = (const float*)d_in[22];
  const float* X2b  = (const float*)d_in[23];
  const float* oW   = (const float*)d_in[24];
  const float* ob   = (const float*)d_in[25];
  float* ws   = (float*)d_ws;
  float* outA = (float*)d_out;                                // (4,16,1024,1024)
  float* out2 = outA + (size_t)NB * AOUT * CSTRIDE;           // (4,1024,64)

  k1_rowstats<<<4096, 256, 0, stream>>>(A, ws);
  k2a_means<<<4, 128, 0, stream>>>(X, ws);
  k2b_R<<<256, 256, 0, stream>>>(X, A_W2, A_W3, A_W4, A_W5, A_W6, A_W7, A_b, ws);
  k2c_xbranch<<<1024, 256, 0, stream>>>(X, X1W1, X1W2, X1W3, X1W4, X1W5, X1W6, X1b,
                                        X2W1, X2W2, X2W3, X2W4, X2W5, X2W6, X2b, ws);
  k3_At<<<512, 256, 65536, stream>>>(A, A_W1, ws, outA);
  k4_out<<<256, 512, 65536, stream>>>(outA, oW, ob, ws, out2);
}